// NetVLAD_18923625906331
// MI455X (gfx1250) — compile-verified
//
#include <hip/hip_runtime.h>

// MI455X / gfx1250: wave32, WMMA 16x16 tiles, TDM async tile staging.
// All matrix math uses V_WMMA_F32_16X16X4_F32 (full fp32, matches the fp32
// reference numerics; the problem is bandwidth/latency bound, so there is no
// reason to drop precision for matrix throughput).

typedef __attribute__((ext_vector_type(2))) float v2f;
typedef __attribute__((ext_vector_type(8))) float v8f;
typedef __attribute__((ext_vector_type(4))) unsigned int v4u;
typedef __attribute__((ext_vector_type(8))) int v8i;
typedef __attribute__((ext_vector_type(4))) int v4i;

#define N_   8
#define C_   256
#define K_   64
#define P_   784          // 28*28
#define PT_  49           // P_/16 pixel tiles
#define F_   (K_*C_)      // 16384 features
#define EPS_ 1e-12f
#define BN_EPS_ 1e-5f

// ---------------------------------------------------------------- utilities
__device__ __forceinline__ v8f wmma_f32(v2f a, v2f b, v8f c) {
    // D = A(16x4) * B(4x16) + C(16x16), fp32
    return __builtin_amdgcn_wmma_f32_16x16x4_f32(
        /*neg_a=*/false, a, /*neg_b=*/false, b,
        /*c_mod=*/(short)0, c, /*reuse_a=*/false, /*reuse_b=*/false);
}

// Tensor Data Mover: DMA a 2D tile (tile_d0 x tile_d1 elements, 4B each) from
// global (row stride stride0 elements) into contiguous LDS at lds_off.
// D# layout per CDNA5 ISA (groups 0/1; 2/3 zero = <=2D).
__device__ __forceinline__ void tdm_load_2d_f32(unsigned lds_off, const void* gptr,
                                                unsigned tile_d0, unsigned tile_d1,
                                                unsigned tens_d0, unsigned tens_d1,
                                                unsigned long long stride0_elems) {
    const unsigned long long ga = (unsigned long long)gptr;
    v4u g0;
    g0.x = 1u;                                        // count=1, user descriptor
    g0.y = lds_off;                                   // lds_addr (bytes)
    g0.z = (unsigned)(ga & 0xffffffffu);              // global_addr[31:0]
    g0.w = (unsigned)((ga >> 32) & 0x01ffffffu)       // global_addr[56:32]
         | (2u << 30);                                // type=2 ("image")
    v8i g1;
    g1[0] = (int)(2u << 16);                          // data_size=2 (4B), mask=0
    g1[1] = (int)((tens_d0 & 0xffffu) << 16);         // tensor_dim0[15:0]
    g1[2] = (int)((tens_d0 >> 16) | ((tens_d1 & 0xffffu) << 16));
    g1[3] = (int)((tens_d1 >> 16) | (tile_d0 << 16)); // tile_dim0
    g1[4] = (int)(tile_d1 & 0xffffu);                 // tile_dim1 (tile_dim2=0)
    g1[5] = (int)(unsigned)(stride0_elems & 0xffffffffu);   // dim0 stride lo32
    g1[6] = (int)(unsigned)((stride0_elems >> 32) & 0xffffu);
    g1[7] = 0;
    const v4i z4 = {0, 0, 0, 0};
#if __clang_major__ >= 23
    const v8i z8 = {0, 0, 0, 0, 0, 0, 0, 0};
    __builtin_amdgcn_tensor_load_to_lds(g0, g1, z4, z4, z8, 0);
#else
    __builtin_amdgcn_tensor_load_to_lds(g0, g1, z4, z4, 0);
#endif
}

// ------------------------------------------------ kernel Z: zero accumulators
__global__ void k_zero(float* Snk, float* nsum) {
    int t = threadIdx.x;
    if (t < N_ * K_) Snk[t] = 0.0f;
    if (t < N_)      nsum[t] = 0.0f;
}

// ------------------------------------------------ kernel A: per-pixel L2 norm
// writes xn twice: [n][c][p] (GEMM1 B operand) and [n][p][c] (GEMM2 B operand)
__global__ void k_normalize(const float* __restrict__ x,
                            float* __restrict__ xn_cp,
                            float* __restrict__ xn_pc) {
    const int n = blockIdx.y;
    const int p = blockIdx.x * blockDim.x + threadIdx.x;
    if (p >= P_) return;
    const float* xin = x + (size_t)n * C_ * P_ + p;
    float ss = 0.0f;
    for (int c = 0; c < C_; ++c) {
        float v = xin[(size_t)c * P_];
        ss += v * v;
    }
    const float inv = 1.0f / fmaxf(sqrtf(ss), EPS_);
    float* ocp = xn_cp + (size_t)n * C_ * P_ + p;
    float* opc = xn_pc + ((size_t)n * P_ + p) * C_;
    for (int c = 0; c < C_; ++c) {
        float v = xin[(size_t)c * P_] * inv;
        ocp[(size_t)c * P_] = v;
        opc[c] = v;
    }
}

// ------------------------------------------------ kernel CC: ||centroid_k||^2
__global__ void k_centnorm(const float* __restrict__ cent, float* __restrict__ cc) {
    int k = threadIdx.x;
    if (k >= K_) return;
    float s = 0.0f;
    for (int c = 0; c < C_; ++c) {
        float v = cent[k * C_ + c];
        s += v * v;
    }
    cc[k] = s;
}

// ------------------------------------------------ kernel B: assign (fused)
// per block: one (n, 16-pixel tile). 4 waves, wave kt owns clusters 16kt..16kt+15.
// xn tile staged by the Tensor Data Mover (wave 0 issues, TENSORcnt handshake).
// dual WMMA accumulators: logits (conv_w) and dots (centroids) over C=256.
// then per-pixel softmax over K=64, rn, w = a/rn, and Snk += w.
__global__ void __launch_bounds__(128)
k_assign(const float* __restrict__ xn_cp,
         const float* __restrict__ conv_w,
         const float* __restrict__ cent,
         const float* __restrict__ cc,
         float* __restrict__ wgt,     // [n][k][p]
         float* __restrict__ Snk) {   // [n][k]
    __shared__ float s_xn[C_ * 16];   // [c][p]  16 KB, filled by TDM
    __shared__ float s_L[K_ * 16];    // logits [k][p]
    __shared__ float s_D[K_ * 16];    // dots   [k][p]
    __shared__ float s_cc[K_];

    const int n  = blockIdx.y;
    const int p0 = blockIdx.x * 16;
    const int tid = threadIdx.x;

    if (tid < 32) {
        // wave 0: DMA the 256x16 tile [c][p] from xn_cp (row stride P_) to LDS
        tdm_load_2d_f32((unsigned)(size_t)(void*)s_xn,
                        xn_cp + (size_t)n * C_ * P_ + p0,
                        /*tile_d0=*/16, /*tile_d1=*/C_,
                        /*tens_d0=*/P_, /*tens_d1=*/C_,
                        /*stride0=*/(unsigned long long)P_);
        __builtin_amdgcn_s_wait_tensorcnt(0);
    }
    if (tid < K_) s_cc[tid] = cc[tid];
    __syncthreads();   // tile visible to all 4 waves

    const int wave    = tid >> 5;         // wave32
    const int lane    = tid & 31;
    const int lane_lo = lane & 15;
    const int lane_hi = lane >> 4;        // selects K-pair / row-half
    const int krow    = wave * 16 + lane_lo;

    v8f accL = {0.f,0.f,0.f,0.f,0.f,0.f,0.f,0.f};
    v8f accD = {0.f,0.f,0.f,0.f,0.f,0.f,0.f,0.f};

    for (int c4 = 0; c4 < C_; c4 += 4) {
        v2f aW = *(const v2f*)(conv_w + krow * C_ + c4 + lane_hi * 2);
        v2f aC = *(const v2f*)(cent   + krow * C_ + c4 + lane_hi * 2);
        v2f b;
        b.x = s_xn[(c4 + lane_hi * 2 + 0) * 16 + lane_lo];
        b.y = s_xn[(c4 + lane_hi * 2 + 1) * 16 + lane_lo];
        accL = wmma_f32(aW, b, accL);
        accD = wmma_f32(aC, b, accD);
    }

    // spill accumulators to LDS:  row = v + 8*lane_hi, col = lane_lo
#pragma unroll
    for (int v = 0; v < 8; ++v) {
        int k = wave * 16 + v + 8 * lane_hi;
        s_L[k * 16 + lane_lo] = accL[v];
        s_D[k * 16 + lane_lo] = accD[v];
    }
    __syncthreads();

    // softmax over K=64 per pixel + intra-norm weight
    if (tid < 16) {
        const int p = tid;
        float mx = -3.0e38f;
        for (int k = 0; k < K_; ++k) mx = fmaxf(mx, s_L[k * 16 + p]);
        float sum = 0.0f;
        for (int k = 0; k < K_; ++k) {
            float e = __expf(s_L[k * 16 + p] - mx);
            s_L[k * 16 + p] = e;
            sum += e;
        }
        const float isum = 1.0f / sum;
        float rn2 = 0.0f;
        for (int k = 0; k < K_; ++k) {
            float a = s_L[k * 16 + p] * isum;
            s_L[k * 16 + p] = a;
            // sum_c (xn_c - cent_kc)^2 = 1 - 2*dot + ||cent_k||^2  (||xn||=1)
            float t = fmaxf(1.0f - 2.0f * s_D[k * 16 + p] + s_cc[k], 0.0f);
            rn2 += a * a * t;
        }
        const float invrn = 1.0f / fmaxf(sqrtf(rn2), EPS_);
        for (int k = 0; k < K_; ++k) {
            float wv = s_L[k * 16 + p] * invrn;
            wgt[((size_t)n * K_ + k) * P_ + p0 + p] = wv;
            atomicAdd(&Snk[n * K_ + k], wv);
        }
    }
}

// ------------------------------------------------ kernel C: VLAD GEMM
// V[n][k][c] = sum_p w[n][k][p] * xn[n][p][c]  -  cent[k][c]*Snk[n][k]
__global__ void __launch_bounds__(128)
k_vlad(const float* __restrict__ wgt,
       const float* __restrict__ xn_pc,
       const float* __restrict__ cent,
       const float* __restrict__ Snk,
       float* __restrict__ vlad) {    // [n][k][c]
    const int n  = blockIdx.y;
    const int c0 = blockIdx.x * 16;
    const int tid = threadIdx.x;
    const int kt      = tid >> 5;       // wave -> k-tile
    const int lane    = tid & 31;
    const int lane_lo = lane & 15;
    const int lane_hi = lane >> 4;
    const int krow    = kt * 16 + lane_lo;

    const float* wrow = wgt + ((size_t)n * K_ + krow) * P_;
    const float* bbase = xn_pc + (size_t)n * P_ * C_;

    v8f acc = {0.f,0.f,0.f,0.f,0.f,0.f,0.f,0.f};
    for (int p4 = 0; p4 < P_; p4 += 4) {
        if (p4 + 4 < P_)  // warm L2/L0 for next B tile -> global_prefetch_b8
            __builtin_prefetch(&bbase[(size_t)(p4 + 4) * C_ + c0], 0, 1);
        v2f a = *(const v2f*)(wrow + p4 + lane_hi * 2);
        v2f b;
        b.x = bbase[(size_t)(p4 + lane_hi * 2 + 0) * C_ + c0 + lane_lo];
        b.y = bbase[(size_t)(p4 + lane_hi * 2 + 1) * C_ + c0 + lane_lo];
        acc = wmma_f32(a, b, acc);
    }
#pragma unroll
    for (int v = 0; v < 8; ++v) {
        int k = kt * 16 + v + 8 * lane_hi;
        int c = c0 + lane_lo;
        float val = acc[v] - cent[k * C_ + c] * Snk[n * K_ + k];
        vlad[((size_t)n * K_ + k) * C_ + c] = val;
    }
}

// ------------------------------------------------ kernel D1: intra-norm over C
__global__ void k_intranorm(float* __restrict__ vlad) {
    __shared__ float red[C_];
    const int nk = blockIdx.x;           // n*K + k
    const int c  = threadIdx.x;
    float v = vlad[(size_t)nk * C_ + c];
    red[c] = v * v;
    __syncthreads();
    for (int s = C_ / 2; s > 0; s >>= 1) {
        if (c < s) red[c] += red[c + s];
        __syncthreads();
    }
    const float inv = 1.0f / fmaxf(sqrtf(red[0]), EPS_);
    vlad[(size_t)nk * C_ + c] = v * inv;
}

// ------------------------------------------------ kernel D2: batchnorm + per-n sumsq
__global__ void k_bn(const float* __restrict__ vlad,
                     const float* __restrict__ bn_w,
                     const float* __restrict__ bn_b,
                     float* __restrict__ out,
                     float* __restrict__ nsum) {
    __shared__ float red[256];
    const int f = blockIdx.x * blockDim.x + threadIdx.x;   // feature = k*C + c
    float v[N_], y[N_];
    float mean = 0.0f;
#pragma unroll
    for (int n = 0; n < N_; ++n) { v[n] = vlad[(size_t)n * F_ + f]; mean += v[n]; }
    mean *= (1.0f / N_);
    float var = 0.0f;
#pragma unroll
    for (int n = 0; n < N_; ++n) { float d = v[n] - mean; var += d * d; }
    var *= (1.0f / N_);                                   // biased, training mode
    const float scale = bn_w[f] * rsqrtf(var + BN_EPS_);
    const float bias  = bn_b[f];
#pragma unroll
    for (int n = 0; n < N_; ++n) {
        y[n] = (v[n] - mean) * scale + bias;
        out[(size_t)n * F_ + f] = y[n];
    }
    // per-image sum of squares for the final L2 norm
    for (int n = 0; n < N_; ++n) {
        red[threadIdx.x] = y[n] * y[n];
        __syncthreads();
        for (int s = 128; s > 0; s >>= 1) {
            if (threadIdx.x < s) red[threadIdx.x] += red[threadIdx.x + s];
            __syncthreads();
        }
        if (threadIdx.x == 0) atomicAdd(&nsum[n], red[0]);
        __syncthreads();
    }
}

// ------------------------------------------------ kernel D3: final L2 scale
__global__ void k_finalnorm(float* __restrict__ out, const float* __restrict__ nsum) {
    const int i = blockIdx.x * blockDim.x + threadIdx.x;   // < N_*F_
    const int n = i / F_;
    out[i] *= 1.0f / fmaxf(sqrtf(nsum[n]), EPS_);
}

// ================================================================ launcher
extern "C" void kernel_launch(void* const* d_in, const int* in_sizes, int n_in,
                              void* d_out, int out_size, void* d_ws, size_t ws_size,
                              hipStream_t stream) {
    const float* x      = (const float*)d_in[0];
    const float* conv_w = (const float*)d_in[1];
    const float* cent   = (const float*)d_in[2];
    const float* bn_w   = (const float*)d_in[3];
    const float* bn_b   = (const float*)d_in[4];
    float* out = (float*)d_out;

    // workspace layout (floats)
    float* ws    = (float*)d_ws;
    float* xn_cp = ws;                          // N*C*P
    float* xn_pc = xn_cp + (size_t)N_*C_*P_;    // N*P*C
    float* wgt   = xn_pc + (size_t)N_*C_*P_;    // N*K*P
    float* Snk   = wgt   + (size_t)N_*K_*P_;    // N*K
    float* cc    = Snk   + (size_t)N_*K_;       // K
    float* vlad  = cc    + (size_t)K_;          // N*K*C
    float* nsum  = vlad  + (size_t)N_*K_*C_;    // N

    k_zero<<<1, 512, 0, stream>>>(Snk, nsum);
    k_normalize<<<dim3((P_ + 255) / 256, N_), 256, 0, stream>>>(x, xn_cp, xn_pc);
    k_centnorm<<<1, 64, 0, stream>>>(cent, cc);
    k_assign<<<dim3(PT_, N_), 128, 0, stream>>>(xn_cp, conv_w, cent, cc, wgt, Snk);
    k_vlad<<<dim3(C_ / 16, N_), 128, 0, stream>>>(wgt, xn_pc, cent, Snk, vlad);
    k_intranorm<<<N_ * K_, C_, 0, stream>>>(vlad);
    k_bn<<<F_ / 256, 256, 0, stream>>>(vlad, bn_w, bn_b, out, nsum);
    k_finalnorm<<<(N_ * F_) / 256, 256, 0, stream>>>(out, nsum);
}